// GraphAttentionLayer_76124000354332
// MI455X (gfx1250) — compile-verified
//
#include <hip/hip_runtime.h>
#include <hip/hip_bf16.h>
#include <cmath>

#define BB 8
#define NN 4096
#define EE 131072
#define FF 64
#define RR (BB * NN)           // 32768 rows of Wh
#define ALPHA 0.2f

typedef __attribute__((ext_vector_type(2))) float v2f;
typedef __attribute__((ext_vector_type(8))) float v8f;

// ---------------------------------------------------------------------------
// Phase 1: Wh = h @ W  via V_WMMA_F32_16X16X4_F32.
// One wave computes a 16x16 tile; K=64 -> 16 WMMA steps.
// A (16x4 f32): lane L holds M=L%16, K = 2*(L/16)+{0,1} in the two VGPRs.
// B (4x16 f32): lane L holds N=L%16, K = 2*(L/16)+{0,1}.
// C/D (16x16):  VGPR v holds M = v + 8*(L/16), N = L%16.
// ---------------------------------------------------------------------------
__global__ __launch_bounds__(256) void k_gemm(const float* __restrict__ h,
                                              const float* __restrict__ W,
                                              float* __restrict__ Wh) {
  const int wave = (blockIdx.x * blockDim.x + threadIdx.x) >> 5;
  const int lane = threadIdx.x & 31;
  const int tile_col = wave & 3;   // 4 column tiles of 16 over F_OUT=64
  const int tile_row = wave >> 2;  // 2048 row tiles of 16 over R=32768
  const int mrow = lane & 15;
  const int khalf = lane >> 4;     // 0 or 1

  const float* __restrict__ hrow = h + (size_t)(tile_row * 16 + mrow) * FF;
  v8f c = {0.f, 0.f, 0.f, 0.f, 0.f, 0.f, 0.f, 0.f};

#pragma unroll
  for (int k0 = 0; k0 < FF; k0 += 4) {
    const int ka = k0 + 2 * khalf;
    v2f a, b;
    a.x = hrow[ka];
    a.y = hrow[ka + 1];
    b.x = W[(size_t)ka * FF + tile_col * 16 + mrow];
    b.y = W[(size_t)(ka + 1) * FF + tile_col * 16 + mrow];
    c = __builtin_amdgcn_wmma_f32_16x16x4_f32(false, a, false, b, (short)0, c,
                                              false, false);
  }

  float* __restrict__ out = Wh + (size_t)(tile_row * 16) * FF + tile_col * 16;
#pragma unroll
  for (int v = 0; v < 8; ++v) {
    out[(size_t)(v + 8 * khalf) * FF + mrow] = c[v];
  }
}

// ---------------------------------------------------------------------------
// Phase 2: per-row score dots s_src = Wh@a1, s_dst = Wh@a2.
// ---------------------------------------------------------------------------
__global__ void k_scores(const float* __restrict__ Wh,
                         const float* __restrict__ a,
                         float* __restrict__ s_src,
                         float* __restrict__ s_dst) {
  const int i = blockIdx.x * blockDim.x + threadIdx.x;
  if (i >= RR) return;
  const float* w = Wh + (size_t)i * FF;
  float acc1 = 0.f, acc2 = 0.f;
#pragma unroll 8
  for (int f = 0; f < FF; ++f) {
    const float v = w[f];
    acc1 += v * a[f];
    acc2 += v * a[FF + f];
  }
  s_src[i] = acc1;
  s_dst[i] = acc2;
}

// ---------------------------------------------------------------------------
// Init: hp=0, denom=0, m=-inf (bit pattern 0xFF800000).
// ---------------------------------------------------------------------------
__global__ void k_init(float* __restrict__ hp, float* __restrict__ denom,
                       float* __restrict__ m) {
  const int i = blockIdx.x * blockDim.x + threadIdx.x;
  if (i < RR * FF) hp[i] = 0.f;
  if (i < RR) {
    denom[i] = 0.f;
    ((unsigned int*)m)[i] = 0xFF800000u;  // -inf
  }
}

__device__ __forceinline__ float edge_e(int b, int e, const int* __restrict__ src,
                                        const int* __restrict__ dst,
                                        const float* __restrict__ ew,
                                        const float* __restrict__ s_src,
                                        const float* __restrict__ s_dst,
                                        int& snode) {
  const int s = src[e];
  const int d = dst[e];
  snode = b * NN + s;
  float v = s_src[snode] + s_dst[b * NN + d];
  v = v > 0.f ? v : ALPHA * v;   // LeakyReLU
  return v * ew[e];
}

// Float atomic max via sign-split int trick (init must be -inf bits).
__device__ __forceinline__ void atomicMaxF32(float* addr, float val) {
  if (!(__float_as_uint(val) >> 31)) {
    atomicMax((int*)addr, __float_as_int(val));
  } else {
    atomicMin((unsigned int*)addr, __float_as_uint(val));
  }
}

// ---------------------------------------------------------------------------
// Phase 3: segment max over src.
// ---------------------------------------------------------------------------
__global__ void k_max(const int* __restrict__ src, const int* __restrict__ dst,
                      const float* __restrict__ ew,
                      const float* __restrict__ s_src,
                      const float* __restrict__ s_dst, float* __restrict__ m) {
  const int t = blockIdx.x * blockDim.x + threadIdx.x;
  if (t >= BB * EE) return;
  const int b = t / EE;
  const int e = t - b * EE;
  int snode;
  const float v = edge_e(b, e, src, dst, ew, s_src, s_dst, snode);
  atomicMaxF32(&m[snode], v);
}

// m = isfinite(m) ? m : 0  (only -inf possible here)
__global__ void k_fixm(float* __restrict__ m) {
  const int i = blockIdx.x * blockDim.x + threadIdx.x;
  if (i >= RR) return;
  const float v = m[i];
  if (__builtin_isinf(v)) m[i] = 0.f;
}

// ---------------------------------------------------------------------------
// Phase 4: denom = segment_sum(exp(e - m[src])).
// ---------------------------------------------------------------------------
__global__ void k_expsum(const int* __restrict__ src, const int* __restrict__ dst,
                         const float* __restrict__ ew,
                         const float* __restrict__ s_src,
                         const float* __restrict__ s_dst,
                         const float* __restrict__ m,
                         float* __restrict__ denom) {
  const int t = blockIdx.x * blockDim.x + threadIdx.x;
  if (t >= BB * EE) return;
  const int b = t / EE;
  const int e = t - b * EE;
  int snode;
  const float v = edge_e(b, e, src, dst, ew, s_src, s_dst, snode);
  atomicAdd(&denom[snode], expf(v - m[snode]));
}

// ---------------------------------------------------------------------------
// Phase 5: hp[b,src,:] += attn * Wh[b,dst,:].  One wave per (b,edge);
// each lane handles 2 of the 64 features. Working set lives in L2.
// ---------------------------------------------------------------------------
__global__ __launch_bounds__(256) void k_scatter(
    const int* __restrict__ src, const int* __restrict__ dst,
    const float* __restrict__ ew, const float* __restrict__ s_src,
    const float* __restrict__ s_dst, const float* __restrict__ m,
    const float* __restrict__ denom, const float* __restrict__ Wh,
    float* __restrict__ hp) {
  const int gid = blockIdx.x * blockDim.x + threadIdx.x;
  const int wave = gid >> 5;
  const int lane = gid & 31;
  if (wave >= BB * EE) return;
  const int b = wave / EE;
  const int e = wave - b * EE;
  int snode;
  const float v = edge_e(b, e, src, dst, ew, s_src, s_dst, snode);
  const float attn = expf(v - m[snode]) / denom[snode];

  const float* __restrict__ whd = Wh + (size_t)(b * NN + dst[e]) * FF;
  float* __restrict__ hps = hp + (size_t)snode * FF;
  const int f = lane * 2;
  atomicAdd(&hps[f],     attn * whd[f]);
  atomicAdd(&hps[f + 1], attn * whd[f + 1]);
}

// ---------------------------------------------------------------------------
// Phase 6: ELU epilogue into d_out (B,N,F) row-major.
// ---------------------------------------------------------------------------
__global__ void k_elu(const float* __restrict__ hp, float* __restrict__ out) {
  const int i = blockIdx.x * blockDim.x + threadIdx.x;
  if (i >= RR * FF) return;
  const float v = hp[i];
  out[i] = v > 0.f ? v : expm1f(v);
}

// ---------------------------------------------------------------------------
extern "C" void kernel_launch(void* const* d_in, const int* in_sizes, int n_in,
                              void* d_out, int out_size, void* d_ws,
                              size_t ws_size, hipStream_t stream) {
  const float* h  = (const float*)d_in[0];           // (B,N,64)
  const int* eidx = (const int*)d_in[1];             // (2,E)
  const int* src  = eidx;
  const int* dst  = eidx + EE;
  const float* ew = (const float*)d_in[2];           // (E,)
  const float* W  = (const float*)d_in[3];           // (64,64)
  const float* a  = (const float*)d_in[4];           // (128,1)
  float* out = (float*)d_out;                        // (B,N,64)

  float* ws    = (float*)d_ws;
  float* Wh    = ws;                 // R*F
  float* hp    = Wh + (size_t)RR * FF;  // R*F
  float* s_src = hp + (size_t)RR * FF;  // R
  float* s_dst = s_src + RR;            // R
  float* m     = s_dst + RR;            // R
  float* denom = m + RR;                // R

  const int T = 256;

  // init hp/denom/m
  k_init<<<(RR * FF + T - 1) / T, T, 0, stream>>>(hp, denom, m);

  // GEMM: (R/16)*(64/16) = 8192 waves -> 262144 threads
  k_gemm<<<(RR / 16) * 4 * 32 / T, T, 0, stream>>>(h, W, Wh);

  // scores
  k_scores<<<(RR + T - 1) / T, T, 0, stream>>>(Wh, a, s_src, s_dst);

  // segment max
  k_max<<<(BB * EE + T - 1) / T, T, 0, stream>>>(src, dst, ew, s_src, s_dst, m);
  k_fixm<<<(RR + T - 1) / T, T, 0, stream>>>(m);

  // segment exp-sum
  k_expsum<<<(BB * EE + T - 1) / T, T, 0, stream>>>(src, dst, ew, s_src, s_dst,
                                                    m, denom);

  // scatter messages: one wave per (b, edge)
  k_scatter<<<(BB * EE * 32) / T, T, 0, stream>>>(src, dst, ew, s_src, s_dst, m,
                                                  denom, Wh, hp);

  // ELU epilogue
  k_elu<<<(RR * FF + T - 1) / T, T, 0, stream>>>(hp, out);
}